// VarPositionalEncoding_58892591563169
// MI455X (gfx1250) — compile-verified
//
#include <hip/hip_runtime.h>

// out[b,v,:] = x[b,v,:] + (index[b,v] >= 0 ? pe[index[b,v],:] : 0)
// x: (1024,128,512) f32, index: (1024,128) i32 (harness narrows i64->i32),
// pe: (2048,512) f32, out: (1024,128,512) f32.
//
// Memory-bound: ~516 MB of HBM traffic -> ~22 us floor at 23.3 TB/s.
// Strategy: B128 vectorized streaming with non-temporal hints for x/out so the
// 4 MB pe table stays resident in the 192 MB L2 (gather becomes L2-hit), plus
// gfx1250 global_prefetch on the x stream.

typedef float v4f __attribute__((ext_vector_type(4)));

#define D_MODEL4 128  // 512 floats per (b,v) row == 128 float4 chunks

__global__ void __launch_bounds__(256) vpe_gather_add_kernel(
    const v4f* __restrict__ x,
    const int* __restrict__ index,
    const v4f* __restrict__ pe,
    v4f*       __restrict__ out,
    int n4)
{
    const int stride = (int)(gridDim.x * blockDim.x);
    int i = (int)(blockIdx.x * blockDim.x + threadIdx.x);

    for (; i < n4; i += stride) {
        // Keep the streaming-load pipe primed (gfx1250 global_prefetch_b8).
        if (i + 2 * stride < n4) {
            __builtin_prefetch(&x[i + 2 * stride], /*rw=*/0, /*locality=*/0);
        }

        // Streamed exactly once: non-temporal so it does not evict pe from L2.
        v4f xv = __builtin_nontemporal_load(&x[i]);

        const int row = i >> 7;              // / D_MODEL4
        const int col = i & (D_MODEL4 - 1);  // % D_MODEL4

        // All 32 lanes of a wave share `row` (128 float4 per row) -> one
        // coalesced dword request per wave.
        const int idx = index[row];

        v4f p = {0.0f, 0.0f, 0.0f, 0.0f};
        if (idx >= 0) {
            // pe is 4 MB: default (RT) temporal hint keeps it L2-resident.
            p = pe[idx * D_MODEL4 + col];
        }

        v4f o;
        o.x = xv.x + p.x;
        o.y = xv.y + p.y;
        o.z = xv.z + p.z;
        o.w = xv.w + p.w;

        // Write-once result: non-temporal store.
        __builtin_nontemporal_store(o, &out[i]);
    }
}

extern "C" void kernel_launch(void* const* d_in, const int* in_sizes, int n_in,
                              void* d_out, int out_size, void* d_ws, size_t ws_size,
                              hipStream_t stream)
{
    const v4f* x     = (const v4f*)d_in[0];   // float32 (1024,128,512)
    const int* index = (const int*)d_in[1];   // int32   (1024,128)
    const v4f* pe    = (const v4f*)d_in[2];   // float32 (2048,512)
    v4f*       out   = (v4f*)d_out;           // float32 (1024,128,512)

    const int n4 = in_sizes[0] / 4;           // 16,777,216 float4 chunks

    const int threads = 256;                  // 8 wave32s per block
    const int iters_per_thread = 8;
    int blocks = (n4 + threads * iters_per_thread - 1) / (threads * iters_per_thread);
    if (blocks < 1) blocks = 1;

    vpe_gather_add_kernel<<<blocks, threads, 0, stream>>>(x, index, pe, out, n4);
}